// Network_35562329211104
// MI455X (gfx1250) — compile-verified
//
#include <hip/hip_runtime.h>
#include <hip/hip_bf16.h>
#include <math.h>

typedef __attribute__((ext_vector_type(2))) float v2f;
typedef __attribute__((ext_vector_type(8))) float v8f;

#define WINDOW 20
#define HOP 10
#define NWIN 399
#define NFRAMES 798   // 2 channels * 399 windows
#define EPSV 1e-6f
#define RIDGEV 0.001f
#define ZSLOT 8000    // guaranteed-zero LDS slot for padded reads

// ---------------------------------------------------------------------------
// Kernel 1: covpool (WMMA f32 Gram) + stem sandwich.  One block per batch b.
//
// Frames are reindexed n -> (c = n&1, win = n>>1): Gram is a frame-order-
// invariant sum, and this makes every gather address a pure increment.
// ---------------------------------------------------------------------------
__global__ __launch_bounds__(256)
void covstem_kernel(const float* __restrict__ x, const float* __restrict__ Wst,
                    float* __restrict__ S0, int B) {
  __shared__ __align__(16) float xs[8004]; // x[b]: [2][4000] + zero slot
  __shared__ float partials[8 * 256];      // per-wave WMMA tile dump
  __shared__ float gram[400];              // 20x20 Gram (f^T f)
  __shared__ float sums[20];               // per-dim frame sums
  __shared__ float covS[400];
  __shared__ float tmpS[400];

  const int b = blockIdx.x;
  const int tid = threadIdx.x;
  const int lane = tid & 31;
  const int wv = tid >> 5;

  // stage x[b] with 128-bit loads
  const float4* xg4 = (const float4*)(x + (size_t)b * 8000);
  float4* xs4 = (float4*)xs;
  for (int i = tid; i < 2000; i += 256) xs4[i] = xg4[i];
  if (tid == 0) xs[ZSLOT] = 0.0f;
  __syncthreads();

  // per-dim sums over all 798 frames (mean correction later)
  if (tid < WINDOW) {
    float s = 0.f;
    for (int c = 0; c < 2; ++c)
      for (int w = 0; w < NWIN; ++w)
        s += xs[c * 4000 + w * HOP + tid];
    sums[tid] = s;
  }

  // ---- WMMA Gram: C(32x32 padded) += A(32x4) * B(4x32), K = 798 ----
  // A 16x4 f32 frag: lanes 0-15 hold M with K={0,1} in vgpr{0,1};
  // lanes 16-31 hold M with K={2,3}.  B mirrors it; since Gram = f^T f the
  // A and B fragments of the same dim-tile are identical registers.
  v8f c00 = {}, c01 = {}, c10 = {}, c11 = {};
  const int m0 = lane & 15;          // dims 0..15 (always valid)
  const int m1 = m0 + 16;            // dims 16..31 (valid iff < 20)
  const bool m1ok = (m1 < WINDOW);
  // frame base for this lane: n0 = step*4 + (lane>>4)*2, step = wv + 8*it
  // n0 even -> channel 0 at win = n0>>1; n0+1 odd -> channel 1, same win.
  const int w0i = wv * 2 + (lane >> 4);
  int a0 = w0i * HOP + m0;                          // f[n0][m0]  (ch 0)
  int a1 = 4000 + w0i * HOP + m0;                   // f[n0+1][m0](ch 1)
  int a2 = m1ok ? (w0i * HOP + m1) : ZSLOT;         // f[n0][m1]
  int a3 = m1ok ? (4000 + w0i * HOP + m1) : ZSLOT;  // f[n0+1][m1]
  const int inc1 = m1ok ? 160 : 0;                  // 32 frames -> +16 wins

  #pragma unroll 4
  for (int it = 0; it < 24; ++it) {  // steps wv, wv+8, ..., wv+184: all valid
    v2f fr0, fr1;
    fr0.x = xs[a0]; fr0.y = xs[a1];
    fr1.x = xs[a2]; fr1.y = xs[a3];
    c00 = __builtin_amdgcn_wmma_f32_16x16x4_f32(false, fr0, false, fr0, (short)0, c00, false, false);
    c01 = __builtin_amdgcn_wmma_f32_16x16x4_f32(false, fr0, false, fr1, (short)0, c01, false, false);
    c10 = __builtin_amdgcn_wmma_f32_16x16x4_f32(false, fr1, false, fr0, (short)0, c10, false, false);
    c11 = __builtin_amdgcn_wmma_f32_16x16x4_f32(false, fr1, false, fr1, (short)0, c11, false, false);
    a0 += 160; a1 += 160; a2 += inc1; a3 += inc1;
  }
  { // tail step = 192 + wv: frames 798/799 (wave 7, lanes 16-31) are padding
    const int n0t = (192 + wv) * 4 + ((lane >> 4) << 1);
    const bool nok = (n0t < NFRAMES);
    const int b0 = nok ? a0 : ZSLOT;
    const int b1 = nok ? a1 : ZSLOT;
    const int b2 = nok ? a2 : ZSLOT;
    const int b3 = nok ? a3 : ZSLOT;
    v2f fr0, fr1;
    fr0.x = xs[b0]; fr0.y = xs[b1];
    fr1.x = xs[b2]; fr1.y = xs[b3];
    c00 = __builtin_amdgcn_wmma_f32_16x16x4_f32(false, fr0, false, fr0, (short)0, c00, false, false);
    c01 = __builtin_amdgcn_wmma_f32_16x16x4_f32(false, fr0, false, fr1, (short)0, c01, false, false);
    c10 = __builtin_amdgcn_wmma_f32_16x16x4_f32(false, fr1, false, fr0, (short)0, c10, false, false);
    c11 = __builtin_amdgcn_wmma_f32_16x16x4_f32(false, fr1, false, fr1, (short)0, c11, false, false);
  }

  // deterministic cross-wave reduction, one 16x16 tile at a time
  for (int tile = 0; tile < 4; ++tile) {
    v8f cc = (tile == 0) ? c00 : (tile == 1) ? c01 : (tile == 2) ? c10 : c11;
    const int ti = tile >> 1, tj = tile & 1;
    __syncthreads();
    for (int v = 0; v < 8; ++v) partials[wv * 256 + v * 32 + lane] = cc[v];
    __syncthreads();
    {
      const int v = tid >> 5, l = tid & 31;
      float s = 0.f;
      for (int w = 0; w < 8; ++w) s += partials[w * 256 + v * 32 + l];
      // C layout: vgpr v holds M=v (lanes 0-15) / M=v+8 (lanes 16-31), N=lane%16
      const int n_t = l & 15, half = l >> 4;
      const int m_t = v + 8 * half;
      const int i = ti * 16 + m_t, j = tj * 16 + n_t;
      if (i < WINDOW && j < WINDOW) gram[i * WINDOW + j] = s;
    }
  }
  __syncthreads();

  // cov = (Gram - N*mu*mu^T)/(N-1) + ridge*I
  for (int idx = tid; idx < 400; idx += 256) {
    const int i = idx / 20, j = idx - (idx / 20) * 20;
    float cv = (gram[idx] - sums[i] * sums[j] * (1.0f / 798.0f)) * (1.0f / 797.0f);
    if (i == j) cv += RIDGEV;
    covS[idx] = cv;
  }
  __syncthreads();

  // S0 = Wst^T * cov * Wst + ridge*I
  for (int idx = tid; idx < 400; idx += 256) {
    const int j = idx / 20, l = idx - j * 20;
    float s = 0.f;
    for (int k = 0; k < 20; ++k) s += covS[j * 20 + k] * Wst[k * 20 + l];
    tmpS[idx] = s;
  }
  __syncthreads();
  for (int idx = tid; idx < 400; idx += 256) {
    const int i = idx / 20, l = idx - i * 20;
    float s = 0.f;
    for (int j = 0; j < 20; ++j) s += Wst[j * 20 + i] * tmpS[j * 20 + l];
    if (i == l) s += RIDGEV;
    S0[(size_t)b * 400 + idx] = s;
  }
}

// ---------------------------------------------------------------------------
// Wave-cooperative cyclic Jacobi eigensolver for symmetric 10x10 in LDS.
// On exit: eigenvalues on diag(A), eigenvectors in columns of V.
// Single-wave workgroup: __syncthreads() is effectively free.
// ---------------------------------------------------------------------------
__device__ void jacobi10(float* A, float* V, int lane) {
  for (int idx = lane; idx < 100; idx += 32)
    V[idx] = ((idx / 10) == (idx - (idx / 10) * 10)) ? 1.0f : 0.0f;
  __syncthreads();
  for (int sweep = 0; sweep < 8; ++sweep) {
    for (int p = 0; p < 9; ++p) {
      for (int q = p + 1; q < 10; ++q) {
        const float apq = A[p * 10 + q];
        const float app = A[p * 10 + p];
        const float aqq = A[q * 10 + q];
        float cth = 1.0f, sth = 0.0f;
        if (fabsf(apq) > 1e-12f) {
          const float tau = (aqq - app) / (2.0f * apq);
          const float t = ((tau >= 0.0f) ? 1.0f : -1.0f) /
                          (fabsf(tau) + sqrtf(1.0f + tau * tau));
          cth = 1.0f / sqrtf(1.0f + t * t);
          sth = t * cth;
        }
        __syncthreads();
        if (lane < 10) { // column rotation on A and V
          const float ajp = A[lane * 10 + p], ajq = A[lane * 10 + q];
          A[lane * 10 + p] = cth * ajp - sth * ajq;
          A[lane * 10 + q] = sth * ajp + cth * ajq;
          const float vjp = V[lane * 10 + p], vjq = V[lane * 10 + q];
          V[lane * 10 + p] = cth * vjp - sth * vjq;
          V[lane * 10 + q] = sth * vjp + cth * vjq;
        }
        __syncthreads();
        if (lane < 10) { // row rotation on A
          const float apj = A[p * 10 + lane], aqj = A[q * 10 + lane];
          A[p * 10 + lane] = cth * apj - sth * aqj;
          A[q * 10 + lane] = sth * apj + cth * aqj;
        }
        __syncthreads();
      }
    }
  }
}

__device__ __forceinline__ void sparsemax6(const float* __restrict__ a, float* w) {
  float z[6];
  for (int i = 0; i < 6; ++i) z[i] = a[i];
  for (int i = 1; i < 6; ++i) { // sort descending
    float key = z[i]; int j = i - 1;
    while (j >= 0 && z[j] < key) { z[j + 1] = z[j]; --j; }
    z[j + 1] = key;
  }
  float csum[6]; float cs = 0.f;
  for (int i = 0; i < 6; ++i) { cs += z[i]; csum[i] = cs; }
  int kz = 0;
  for (int i = 0; i < 6; ++i) kz += (1.0f + (float)(i + 1) * z[i] > csum[i]) ? 1 : 0;
  const float tau = (csum[kz - 1] - 1.0f) / (float)kz;
  for (int i = 0; i < 6; ++i) w[i] = fmaxf(a[i] - tau, 0.0f);
}

// ---------------------------------------------------------------------------
// Kernel 2: one SPD cell.  One wave per (m,b,c) item.
//   cand_k = W_k^T S W_k + ridge I ; L_k = logm(cand_k)
//   out[b, m*Cin+c] = expm( sum_k sparsemax(alpha)_k * L_k )
// ---------------------------------------------------------------------------
__global__ __launch_bounds__(32)
void cell_kernel(const float* __restrict__ Sin, const float* __restrict__ Wop,
                 const float* __restrict__ alpha, float* __restrict__ Sout,
                 int B, int Cin, int din) {
  __shared__ float Sw[400];
  __shared__ float Tm[200];
  __shared__ float Am[100];
  __shared__ float Vm[100];
  __shared__ float Ts[100];

  const int lane = threadIdx.x;
  const int item = blockIdx.x;
  const int b = item / (2 * Cin);
  const int rest = item - b * (2 * Cin);
  const int m = rest / Cin;
  const int c = rest - m * Cin;

  const float* Sp = Sin + ((size_t)b * Cin + c) * din * din;
  for (int idx = lane; idx < din * din; idx += 32) Sw[idx] = Sp[idx];

  float w[6];
  sparsemax6(alpha, w);

  float acc[10];
  for (int i = 0; i < 10; ++i) acc[i] = 0.f;
  __syncthreads();

  for (int k = 0; k < 6; ++k) {
    const float* Wk = Wop + (size_t)(m * 6 + k) * din * 10; // [din][10]
    // Tm = S @ Wk   (din x 10)
    for (int idx = lane; idx < din * 10; idx += 32) {
      const int j = idx / 10, n = idx - j * 10;
      float s = 0.f;
      for (int l = 0; l < din; ++l) s += Sw[j * din + l] * Wk[l * 10 + n];
      Tm[idx] = s;
    }
    __syncthreads();
    // Am = Wk^T @ Tm + ridge I   (10 x 10)
    for (int idx = lane; idx < 100; idx += 32) {
      const int i = idx / 10, n = idx - (idx / 10) * 10;
      float s = 0.f;
      for (int j = 0; j < din; ++j) s += Wk[j * 10 + i] * Tm[j * 10 + n];
      if (i == n) s += RIDGEV;
      Am[idx] = s;
    }
    __syncthreads();
    jacobi10(Am, Vm, lane);
    // acc += w[k] * V diag(log(max(ev,eps))) V^T   (lane j owns column j)
    if (lane < 10) {
      for (int d = 0; d < 10; ++d) {
        const float lw = logf(fmaxf(Am[d * 10 + d], EPSV));
        const float cj = w[k] * lw * Vm[lane * 10 + d];
        for (int i = 0; i < 10; ++i) acc[i] += cj * Vm[i * 10 + d];
      }
    }
    __syncthreads();
  }

  // symmetrize accumulated log-matrix, then expm
  if (lane < 10)
    for (int i = 0; i < 10; ++i) Ts[i * 10 + lane] = acc[i];
  __syncthreads();
  for (int idx = lane; idx < 100; idx += 32) {
    const int i = idx / 10, j = idx - i * 10;
    Am[idx] = 0.5f * (Ts[idx] + Ts[j * 10 + i]);
  }
  __syncthreads();
  jacobi10(Am, Vm, lane);
  if (lane < 10) {
    float g[10];
    for (int i = 0; i < 10; ++i) g[i] = 0.f;
    for (int d = 0; d < 10; ++d) {
      const float ew = expf(Am[d * 10 + d]);
      const float cj = ew * Vm[lane * 10 + d];
      for (int i = 0; i < 10; ++i) g[i] += cj * Vm[i * 10 + d];
    }
    const int cout = m * Cin + c;
    float* dst = Sout + ((size_t)b * (2 * Cin) + cout) * 100;
    for (int i = 0; i < 10; ++i) dst[i * 10 + lane] = g[i];
  }
}

// ---------------------------------------------------------------------------
// Kernel 3: final logm of S (B,8,10,10) -> Lflat (B,800)
// ---------------------------------------------------------------------------
__global__ __launch_bounds__(32)
void logm_kernel(const float* __restrict__ Sin, float* __restrict__ Lf, int B) {
  __shared__ float Am[100];
  __shared__ float Vm[100];
  const int lane = threadIdx.x;
  const int b = blockIdx.x >> 3;
  const int ch = blockIdx.x & 7;
  const float* Sp = Sin + ((size_t)b * 8 + ch) * 100;
  for (int idx = lane; idx < 100; idx += 32) {
    const int i = idx / 10, j = idx - i * 10;
    Am[idx] = 0.5f * (Sp[idx] + Sp[j * 10 + i]);
  }
  __syncthreads();
  jacobi10(Am, Vm, lane);
  if (lane < 10) {
    float g[10];
    for (int i = 0; i < 10; ++i) g[i] = 0.f;
    for (int d = 0; d < 10; ++d) {
      const float lw = logf(fmaxf(Am[d * 10 + d], EPSV));
      const float cj = lw * Vm[lane * 10 + d];
      for (int i = 0; i < 10; ++i) g[i] += cj * Vm[i * 10 + d];
    }
    float* dst = Lf + (size_t)b * 800 + ch * 100;
    for (int i = 0; i < 10; ++i) dst[i * 10 + lane] = g[i];
  }
}

// ---------------------------------------------------------------------------
// Kernel 4: classifier  out[b,n] = Lf[b,:] . Wc[:,n] + bc[n]
// ---------------------------------------------------------------------------
__global__ __launch_bounds__(256)
void cls_kernel(const float* __restrict__ Lf, const float* __restrict__ Wc,
                const float* __restrict__ bc, float* __restrict__ out, int B) {
  const int gid = blockIdx.x * 256 + threadIdx.x;
  if (gid >= B * 10) return;
  const int b = gid / 10, n = gid - b * 10;
  float s = bc[n];
  const float* lr = Lf + (size_t)b * 800;
  for (int t = 0; t < 800; ++t) s += lr[t] * Wc[t * 10 + n];
  out[gid] = s;
}

// ---------------------------------------------------------------------------
extern "C" void kernel_launch(void* const* d_in, const int* in_sizes, int n_in,
                              void* d_out, int out_size, void* d_ws, size_t ws_size,
                              hipStream_t stream) {
  const float* x    = (const float*)d_in[0];
  const float* Wst  = (const float*)d_in[1];
  const float* Wop0 = (const float*)d_in[2];
  const float* Wop1 = (const float*)d_in[3];
  const float* Wop2 = (const float*)d_in[4];
  const float* alph = (const float*)d_in[5];
  const float* Wcls = (const float*)d_in[6];
  const float* bcls = (const float*)d_in[7];
  float* out = (float*)d_out;

  const int B = in_sizes[0] / (2 * 4000); // 2048

  // workspace layout (floats): S0 | SA | SB | Lflat  (~23 MB total)
  float* ws = (float*)d_ws;
  float* S0 = ws;                        // B*400
  float* SA = S0 + (size_t)B * 400;      // B*800 (8 ch * 100 max)
  float* SB = SA + (size_t)B * 800;      // B*800
  float* Lf = SB + (size_t)B * 800;      // B*800

  covstem_kernel<<<B, 256, 0, stream>>>(x, Wst, S0, B);
  cell_kernel<<<2 * B * 1, 32, 0, stream>>>(S0, Wop0, alph + 0,  SA, B, 1, 20);
  cell_kernel<<<2 * B * 2, 32, 0, stream>>>(SA, Wop1, alph + 6,  SB, B, 2, 10);
  cell_kernel<<<2 * B * 4, 32, 0, stream>>>(SB, Wop2, alph + 12, SA, B, 4, 10);
  logm_kernel<<<B * 8, 32, 0, stream>>>(SA, Lf, B);
  cls_kernel<<<(B * 10 + 255) / 256, 256, 0, stream>>>(Lf, Wcls, bcls, out, B);
}